// FeatureDecorr_2628519985843
// MI455X (gfx1250) — compile-verified
//
#include <hip/hip_runtime.h>
#include <stdint.h>

typedef float v2f __attribute__((ext_vector_type(2)));
typedef float v8f __attribute__((ext_vector_type(8)));
typedef int   i4v __attribute__((ext_vector_type(4)));

#define NBATCH 32
#define CCH    512
#define GRP    64
#define HWSZ   2304      // 48*48
#define MTOT   18432     // 8*2304 samples per group
#define EPSV   1e-5f
#define NITER  10

#define CHUNK      96
#define CH_PER_HW  24    // 2304/96
#define NCHUNK     192   // chunks per batch image (8 kk-blocks * 24)
#define C4PC       24    // float4s per row per chunk
#define XS_PAD     100   // 96+4 pad: stride%64==36 -> conflict-free column reads
#define DS_PAD     68
#define NS_PAD     65
#define STATS_BLKS 16    // blocks per n in k_stats -> 12 chunks each

// ------------------------------------------------- CDNA5 async-to-LDS copy
__device__ __forceinline__ void async_ld_b128(const void* gptr, void* lptr) {
#if defined(__has_builtin) && __has_builtin(__builtin_amdgcn_global_load_async_to_lds_b128)
  __builtin_amdgcn_global_load_async_to_lds_b128((i4v*)gptr, (i4v*)lptr, 0, 0);
#else
  const unsigned l = (unsigned)(uintptr_t)lptr;
  asm volatile("global_load_async_to_lds_b128 %0, %1, off"
               :: "v"(l), "v"(gptr) : "memory");
#endif
}

__device__ __forceinline__ void wait_async0() {
#if defined(__has_builtin) && __has_builtin(__builtin_amdgcn_s_wait_asynccnt)
  __builtin_amdgcn_s_wait_asynccnt(0);
#else
  asm volatile("s_wait_asynccnt 0x0" ::: "memory");
#endif
}

// ---------------------------------------------------------------- utilities
__global__ void k_zero(float* __restrict__ p, int n) {
  int i = blockIdx.x * blockDim.x + threadIdx.x;
  if (i < n) p[i] = 0.0f;
}

// ---------------------------------------------------------------- kernel 1:
// raw Gram R[n] = X_n X_n^T (64x64, K=18432) + row sums S[n][g].
// Double-buffered async global->LDS staging overlapped with f32 WMMA.
__global__ __launch_bounds__(256) void k_stats(const float* __restrict__ x,
                                               float* __restrict__ R,
                                               float* __restrict__ S) {
  __shared__ __align__(16) float xs[2][GRP][XS_PAD];

  const int tid  = threadIdx.x;
  const int lane = tid & 31;
  const int wv   = tid >> 5;       // 8 waves
  const int hi   = lane >> 4;
  const int lr   = lane & 15;
  const int k2   = hi * 2;         // f32 A/B fragment: lanes16-31 hold K+2
  const int n    = blockIdx.y;

  // staging assignment: 4 threads per row, 6 float4 each
  const int srow4 = tid >> 2;
  const int sq    = tid & 3;

  // each wave owns 2 of 16 Gram tiles (4x4 tiling of 64x64)
  const int t0 = wv * 2, t1 = wv * 2 + 1;
  const int i0 = t0 >> 2, j0 = t0 & 3;
  const int i1 = t1 >> 2, j1 = t1 & 3;
  v8f acc0 = {};
  v8f acc1 = {};

  const int srow = tid & 63;       // row-sum ownership: 4 threads per row
  const int scol = (tid >> 6) * C4PC;
  float ssum = 0.0f;

  const float* xn = x + (size_t)n * CCH * HWSZ;

  // prologue: stage first chunk into buffer 0
  {
    const int ch    = blockIdx.x;
    const int kk    = ch / CH_PER_HW;
    const int hwoff = (ch - kk * CH_PER_HW) * CHUNK;
    const float* base = xn + (size_t)kk * GRP * HWSZ + hwoff;
#pragma unroll
    for (int s = 0; s < 6; ++s) {
      const int c4 = sq * 6 + s;
      async_ld_b128(base + (size_t)srow4 * HWSZ + c4 * 4, &xs[0][srow4][c4 * 4]);
    }
  }
  wait_async0();
  __syncthreads();

  int buf = 0;
  for (int ch = blockIdx.x; ch < NCHUNK; ch += STATS_BLKS) {
    // stage next chunk into the other buffer (overlapped with compute)
    const int nxt = ch + STATS_BLKS;
    if (nxt < NCHUNK) {
      const int kk    = nxt / CH_PER_HW;
      const int hwoff = (nxt - kk * CH_PER_HW) * CHUNK;
      const float* base = xn + (size_t)kk * GRP * HWSZ + hwoff;
#pragma unroll
      for (int s = 0; s < 6; ++s) {
        const int c4 = sq * 6 + s;
        async_ld_b128(base + (size_t)srow4 * HWSZ + c4 * 4,
                      &xs[buf ^ 1][srow4][c4 * 4]);
      }
    }

    // row-sum partials (for the mean)
    {
      float s = 0.0f;
#pragma unroll
      for (int c = 0; c < CHUNK / 4; ++c) s += xs[buf][srow][scol + c];
      ssum += s;
    }

    // Gram accumulation: K loop over the 96-slab, f32 WMMA 16x16x4
    for (int k = 0; k < CHUNK; k += 4) {
      v2f a0, b0, a1, b1;
      a0.x = xs[buf][i0 * 16 + lr][k + k2];
      a0.y = xs[buf][i0 * 16 + lr][k + k2 + 1];
      b0.x = xs[buf][j0 * 16 + lr][k + k2];
      b0.y = xs[buf][j0 * 16 + lr][k + k2 + 1];
      acc0 = __builtin_amdgcn_wmma_f32_16x16x4_f32(false, a0, false, b0,
                                                   (short)0, acc0, false, false);
      a1.x = xs[buf][i1 * 16 + lr][k + k2];
      a1.y = xs[buf][i1 * 16 + lr][k + k2 + 1];
      b1.x = xs[buf][j1 * 16 + lr][k + k2];
      b1.y = xs[buf][j1 * 16 + lr][k + k2 + 1];
      acc1 = __builtin_amdgcn_wmma_f32_16x16x4_f32(false, a1, false, b1,
                                                   (short)0, acc1, false, false);
    }

    wait_async0();      // next buffer's async transfers (this wave) done
    __syncthreads();    // everyone done reading buf + everyone's asyncs done
    buf ^= 1;
  }

  atomicAdd(&S[n * GRP + srow], ssum);

  float* Rn = R + (size_t)n * GRP * GRP;
#pragma unroll
  for (int v = 0; v < 8; ++v) {
    atomicAdd(&Rn[(i0 * 16 + v + hi * 8) * GRP + j0 * 16 + lr], acc0[v]);
    atomicAdd(&Rn[(i1 * 16 + v + hi * 8) * GRP + j1 * 16 + lr], acc1[v]);
  }
}

// ---------------------------------------------------------------- kernel 2:
// Newton-Schulz isqrt of 64x64 cov (one block per n), matmuls via WMMA.
__device__ __forceinline__ void mm64_wmma(const float* __restrict__ A,
                                          const float* __restrict__ B,
                                          v8f out2[2], int tid) {
  const int lane = tid & 31, wv = tid >> 5;
  const int hi = lane >> 4, lr = lane & 15, k2 = hi * 2;
#pragma unroll
  for (int q = 0; q < 2; ++q) {
    const int t = wv * 2 + q;
    const int i = t >> 2, j = t & 3;
    v8f acc = {};
#pragma unroll
    for (int k = 0; k < GRP; k += 4) {
      v2f a, b;
      a.x = A[(i * 16 + lr) * NS_PAD + k + k2];
      a.y = A[(i * 16 + lr) * NS_PAD + k + k2 + 1];
      b.x = B[(k + k2) * NS_PAD + j * 16 + lr];
      b.y = B[(k + k2 + 1) * NS_PAD + j * 16 + lr];
      acc = __builtin_amdgcn_wmma_f32_16x16x4_f32(false, a, false, b,
                                                  (short)0, acc, false, false);
    }
    out2[q] = acc;
  }
}

__global__ __launch_bounds__(256) void k_newton_schulz(
    const float* __restrict__ R, const float* __restrict__ S,
    float* __restrict__ D, float* __restrict__ dm) {
  __shared__ float Yb[GRP * NS_PAD];
  __shared__ float Zb[GRP * NS_PAD];
  __shared__ float Tb[GRP * NS_PAD];
  __shared__ float meanv[GRP];
  __shared__ float redbuf;

  const int tid  = threadIdx.x;
  const int n    = blockIdx.x;
  const int lane = tid & 31, wv = tid >> 5;
  const int hi = lane >> 4, lr = lane & 15;

  if (tid < GRP) meanv[tid] = S[n * GRP + tid] * (1.0f / (float)MTOT);
  if (tid == 0) redbuf = 0.0f;
  __syncthreads();

  // cov = R/M - mu mu^T + eps I (built in Tb), plus Frobenius norm^2
  const float* Rn = R + (size_t)n * GRP * GRP;
  float lsq = 0.0f;
  for (int e = tid; e < GRP * GRP; e += 256) {
    const int g = e >> 6, h = e & 63;
    float c = Rn[e] * (1.0f / (float)MTOT) - meanv[g] * meanv[h];
    if (g == h) c += EPSV;
    Tb[g * NS_PAD + h] = c;
    lsq += c * c;
  }
  atomicAdd(&redbuf, lsq);
  __syncthreads();

  const float fro = sqrtf(redbuf);
  const float inv = 1.0f / fro;
  for (int e = tid; e < GRP * GRP; e += 256) {
    const int g = e >> 6, h = e & 63;
    Yb[g * NS_PAD + h] = Tb[g * NS_PAD + h] * inv;
    Zb[g * NS_PAD + h] = (g == h) ? 1.0f : 0.0f;
  }
  __syncthreads();

  v8f o[2];
  for (int it = 0; it < NITER; ++it) {
    // T = 1.5 I - 0.5 (Z*Y)   (distinct buffer -> direct store)
    mm64_wmma(Zb, Yb, o, tid);
#pragma unroll
    for (int q = 0; q < 2; ++q) {
      const int t = wv * 2 + q, i = t >> 2, j = t & 3;
#pragma unroll
      for (int v = 0; v < 8; ++v) {
        const int r = i * 16 + v + hi * 8;
        const int c = j * 16 + lr;
        Tb[r * NS_PAD + c] = ((r == c) ? 1.5f : 0.0f) - 0.5f * o[q][v];
      }
    }
    __syncthreads();
    // Y = Y*T (register-staged in-place update)
    mm64_wmma(Yb, Tb, o, tid);
    __syncthreads();
#pragma unroll
    for (int q = 0; q < 2; ++q) {
      const int t = wv * 2 + q, i = t >> 2, j = t & 3;
#pragma unroll
      for (int v = 0; v < 8; ++v)
        Yb[(i * 16 + v + hi * 8) * NS_PAD + j * 16 + lr] = o[q][v];
    }
    __syncthreads();
    // Z = T*Z (register-staged in-place update)
    mm64_wmma(Tb, Zb, o, tid);
    __syncthreads();
#pragma unroll
    for (int q = 0; q < 2; ++q) {
      const int t = wv * 2 + q, i = t >> 2, j = t & 3;
#pragma unroll
      for (int v = 0; v < 8; ++v)
        Zb[(i * 16 + v + hi * 8) * NS_PAD + j * 16 + lr] = o[q][v];
    }
    __syncthreads();
  }

  const float dscale = rsqrtf(fro);   // D = Z / sqrt(||A||_F)
  float* Dn = D + (size_t)n * GRP * GRP;
  for (int e = tid; e < GRP * GRP; e += 256) {
    const int g = e >> 6, h = e & 63;
    Dn[e] = Zb[g * NS_PAD + h] * dscale;
  }
  if (tid < GRP) {               // dm = D * mu (folds mean subtraction)
    float a = 0.0f;
    for (int h = 0; h < GRP; ++h) a += Zb[tid * NS_PAD + h] * meanv[h];
    dm[n * GRP + tid] = a * dscale;
  }
}

// ---------------------------------------------------------------- kernel 3:
// y = D*x - dm, fused weight/bias; async staging, 64x96 output block via WMMA.
__global__ __launch_bounds__(256) void k_apply(const float* __restrict__ x,
                                               const float* __restrict__ D,
                                               const float* __restrict__ dm,
                                               const float* __restrict__ weight,
                                               const float* __restrict__ bias,
                                               float* __restrict__ out) {
  __shared__ __align__(16) float ds[GRP][DS_PAD];
  __shared__ __align__(16) float xs[GRP][XS_PAD];
  __shared__ float dms[GRP], wvv[GRP], bvv[GRP];

  const int tid  = threadIdx.x;
  const int lane = tid & 31, wave = tid >> 5;
  const int hi = lane >> 4, lr = lane & 15, k2 = hi * 2;
  const int n  = blockIdx.y;
  const int ch = blockIdx.x;
  const int kk    = ch / CH_PER_HW;
  const int hwoff = (ch - kk * CH_PER_HW) * CHUNK;

  const int srow4 = tid >> 2;
  const int sq    = tid & 3;

  // async-stage D[n] (64x64) and the 64x96 x slab into LDS
  const float* Dn = D + (size_t)n * GRP * GRP;
#pragma unroll
  for (int s = 0; s < 4; ++s) {
    const int c4 = sq * 4 + s;                       // 16 float4 per row
    async_ld_b128(Dn + (size_t)srow4 * GRP + c4 * 4, &ds[srow4][c4 * 4]);
  }
  const float* base = x + ((size_t)n * CCH + (size_t)kk * GRP) * HWSZ + hwoff;
#pragma unroll
  for (int s = 0; s < 6; ++s) {
    const int c4 = sq * 6 + s;                       // 24 float4 per row
    async_ld_b128(base + (size_t)srow4 * HWSZ + c4 * 4, &xs[srow4][c4 * 4]);
  }
  if (tid < GRP) {
    dms[tid] = dm[n * GRP + tid];
    wvv[tid] = weight[kk * GRP + tid];
    bvv[tid] = bias[kk * GRP + tid];
  }
  wait_async0();
  __syncthreads();

  float* obase = out + ((size_t)n * CCH + (size_t)kk * GRP) * HWSZ + hwoff;

  // 24 output tiles (4 g-tiles x 6 m-tiles), 3 per wave; K = 64
#pragma unroll
  for (int q = 0; q < 3; ++q) {
    const int t = wave * 3 + q;
    const int i = t / 6, j = t % 6;
    v8f acc = {};
#pragma unroll
    for (int k = 0; k < GRP; k += 4) {
      v2f a, b;
      a.x = ds[i * 16 + lr][k + k2];
      a.y = ds[i * 16 + lr][k + k2 + 1];
      b.x = xs[k + k2][j * 16 + lr];
      b.y = xs[k + k2 + 1][j * 16 + lr];
      acc = __builtin_amdgcn_wmma_f32_16x16x4_f32(false, a, false, b,
                                                  (short)0, acc, false, false);
    }
#pragma unroll
    for (int v = 0; v < 8; ++v) {
      const int g   = i * 16 + v + hi * 8;
      const int col = j * 16 + lr;
      obase[(size_t)g * HWSZ + col] = (acc[v] - dms[g]) * wvv[g] + bvv[g];
    }
  }
}

// ---------------------------------------------------------------- launcher
extern "C" void kernel_launch(void* const* d_in, const int* in_sizes, int n_in,
                              void* d_out, int out_size, void* d_ws,
                              size_t ws_size, hipStream_t stream) {
  const float* x = (const float*)d_in[0];
  const float* w = (const float*)d_in[1];
  const float* b = (const float*)d_in[2];
  float* out = (float*)d_out;

  float* ws = (float*)d_ws;
  float* R  = ws;                 // 32*64*64 = 131072 f32
  float* S  = ws + 131072;        // 32*64    =   2048 f32
  float* D  = ws + 133120;        // 32*64*64 = 131072 f32
  float* dm = ws + 264192;        // 32*64    =   2048 f32
  (void)in_sizes; (void)n_in; (void)out_size; (void)ws_size;

  k_zero<<<dim3((133120 + 255) / 256), dim3(256), 0, stream>>>(ws, 133120);
  k_stats<<<dim3(STATS_BLKS, NBATCH), dim3(256), 0, stream>>>(x, R, S);
  k_newton_schulz<<<dim3(NBATCH), dim3(256), 0, stream>>>(R, S, D, dm);
  k_apply<<<dim3(NCHUNK, NBATCH), dim3(256), 0, stream>>>(x, D, dm, w, b, out);
}